// BGA_25357486916128
// MI455X (gfx1250) — compile-verified
//
#include <hip/hip_runtime.h>
#include <hip/hip_bf16.h>
#include <climits>

// ---------------------------------------------------------------------------
// BGA GNN forward for MI455X (gfx1250, wave32).
// GEMMs use V_WMMA_F32_16X16X4_F32 (full fp32: GEMM flops ~10GF are trivial;
// the workload is bound by ~1GB of edge-gather/scatter traffic, ~50us @23.3TB/s,
// and N*256 fp32 node tensors (20MB) fit in the 192MB L2, so gathers hit L2).
// Each wave computes a 16x64 output strip: one A fragment feeds 4 WMMAs.
// ---------------------------------------------------------------------------

#define Nn    20000
#define Ee    320000
#define F_IN  64
#define HID   128
#define NH    2
#define Gg    64
#define MIDd  32

typedef float v2f __attribute__((ext_vector_type(2)));
typedef float v8f __attribute__((ext_vector_type(8)));

#define WMMA_F32(a, b, c) \
  __builtin_amdgcn_wmma_f32_16x16x4_f32(false, (a), false, (b), (short)0, (c), false, false)

// ---------------- fp32 WMMA GEMM:  C[N x M] = A[N x K] @ W[K x M] + bias
// one wave per 16(rows) x 64(cols) strip; K stepped by 4; M % 64 == 0.
// A 16x4 layout : lanes 0-15 -> M=lane, v0=K0,v1=K1 ; lanes 16-31 -> K2,K3
// B 4x16 layout : lanes 0-15 -> N=lane, v0=K0,v1=K1 ; lanes 16-31 -> K2,K3
// C 16x16 layout: vgpr r: lanes 0-15 -> M=r, lanes 16-31 -> M=r+8 ; N=lane&15
__global__ void gemm_bias_wmma_f32(const float* __restrict__ A,
                                   const float* __restrict__ W,
                                   const float* __restrict__ bias,
                                   float* __restrict__ C,
                                   int K, int M) {
  const int col0 = blockIdx.x * 64;
  const int row0 = blockIdx.y * 16;
  const int lane = threadIdx.x;
  const int half = lane >> 4;     // K half selector
  const int l16  = lane & 15;

  v8f acc0 = {}, acc1 = {}, acc2 = {}, acc3 = {};
  const float* __restrict__ arow  = A + (size_t)(row0 + l16) * K + 2 * half;
  const float* __restrict__ wbase = W + col0 + l16 + (size_t)(2 * half) * M;
  for (int k0 = 0; k0 < K; k0 += 4) {
    v2f a = *(const v2f*)(arow + k0);            // A[row][k0+2h .. +1]
    const float* wk = wbase + (size_t)k0 * M;
    v2f b0, b1, b2, b3;
    b0.x = wk[0];  b0.y = wk[M];                 // W[k0+2h][col+ 0], W[k0+2h+1][col+ 0]
    b1.x = wk[16]; b1.y = wk[M + 16];
    b2.x = wk[32]; b2.y = wk[M + 32];
    b3.x = wk[48]; b3.y = wk[M + 48];
    acc0 = WMMA_F32(a, b0, acc0);
    acc1 = WMMA_F32(a, b1, acc1);
    acc2 = WMMA_F32(a, b2, acc2);
    acc3 = WMMA_F32(a, b3, acc3);
  }
  float* __restrict__ crow = C + (size_t)(row0 + 8 * half) * M + col0 + l16;
  const float bv0 = bias[col0 + l16];
  const float bv1 = bias[col0 + 16 + l16];
  const float bv2 = bias[col0 + 32 + l16];
  const float bv3 = bias[col0 + 48 + l16];
#pragma unroll
  for (int r = 0; r < 8; ++r) {
    float* p = crow + (size_t)r * M;
    p[0]  = acc0[r] + bv0;
    p[16] = acc1[r] + bv1;
    p[32] = acc2[r] + bv2;
    p[48] = acc3[r] + bv3;
  }
}

// ---------------- elementwise / reduction helpers ----------------
__global__ void fill_i32(int* p, int v, int n) {
  int i = blockIdx.x * blockDim.x + threadIdx.x;
  if (i < n) p[i] = v;
}

// acc[row[e]*C + c] += x[col[e]*C + c]
__global__ void edge_gather_add(const float* __restrict__ x,
                                const int* __restrict__ rows,
                                const int* __restrict__ cols,
                                float* __restrict__ acc, int total, int C) {
  int i = blockIdx.x * blockDim.x + threadIdx.x;
  if (i >= total) return;
  int e = i / C, c = i - e * C;
  unsafeAtomicAdd(&acc[(size_t)rows[e] * C + c], x[(size_t)cols[e] * C + c]);
}

// per-channel training-mode BN stats over N rows (one block per channel)
__global__ void bn_stats(const float* __restrict__ y, int C,
                         float* __restrict__ mean, float* __restrict__ inv) {
  const int c = blockIdx.x;
  float s = 0.f, q = 0.f;
  for (int r = threadIdx.x; r < Nn; r += blockDim.x) {
    float v = y[(size_t)r * C + c];
    s += v; q += v * v;
  }
  __shared__ float ss[256], sq[256];
  ss[threadIdx.x] = s; sq[threadIdx.x] = q;
  __syncthreads();
  for (int o = blockDim.x >> 1; o > 0; o >>= 1) {
    if (threadIdx.x < o) { ss[threadIdx.x] += ss[threadIdx.x + o];
                           sq[threadIdx.x] += sq[threadIdx.x + o]; }
    __syncthreads();
  }
  if (threadIdx.x == 0) {
    float m = ss[0] / (float)Nn;
    float v = sq[0] / (float)Nn - m * m;
    mean[c] = m;
    inv[c]  = rsqrtf(v + 1e-5f);
  }
}

__global__ void bn_relu(const float* __restrict__ y,
                        const float* __restrict__ mean, const float* __restrict__ inv,
                        const float* __restrict__ g, const float* __restrict__ b,
                        float* __restrict__ out, int n, int C) {
  int i = blockIdx.x * blockDim.x + threadIdx.x;
  if (i >= n) return;
  int c = i % C;
  float v = (y[i] - mean[c]) * inv[c] * g[c] + b[c];
  out[i] = v > 0.f ? v : 0.f;
}

__global__ void bias_relu(const float* __restrict__ acc, const float* __restrict__ bias,
                          float* __restrict__ out, int n, int C) {
  int i = blockIdx.x * blockDim.x + threadIdx.x;
  if (i >= n) return;
  int c = i % C;
  float v = acc[i] + bias[c];
  out[i] = v > 0.f ? v : 0.f;
}

// ---------------- attention (GATv2-style) ----------------
// logit[e,h] = sum_c leaky_relu(xl[src,h,c]+xr[dst,h,c], 0.2) * att[h,c]
// + running float-max per (dst,h) via monotone int transform.
__global__ void att_logits(const float* __restrict__ xl, const float* __restrict__ xr,
                           const int* __restrict__ src, const int* __restrict__ dst,
                           const float* __restrict__ att,
                           float* __restrict__ logits, int* __restrict__ lmaxi) {
  int i = blockIdx.x * blockDim.x + threadIdx.x;
  if (i >= Ee * NH) return;
  int e = i >> 1, h = i & 1;
  const float* pl = xl + (size_t)src[e] * (NH * HID) + h * HID;
  const float* pr = xr + (size_t)dst[e] * (NH * HID) + h * HID;
  const float* pa = att + h * HID;
  float s = 0.f;
#pragma unroll 4
  for (int c = 0; c < HID; ++c) {
    float v = pl[c] + pr[c];
    v = v > 0.f ? v : 0.2f * v;
    s += v * pa[c];
  }
  logits[i] = s;
  int key = __float_as_int(s);
  key ^= (key >> 31) & 0x7FFFFFFF;           // monotone float -> signed int
  atomicMax(&lmaxi[(size_t)dst[e] * NH + h], key);
}

__global__ void att_expw(const float* __restrict__ logits, const int* __restrict__ lmaxi,
                         const int* __restrict__ dst,
                         float* __restrict__ wbuf, float* __restrict__ denom) {
  int i = blockIdx.x * blockDim.x + threadIdx.x;
  if (i >= Ee * NH) return;
  int e = i >> 1, h = i & 1;
  int lm = lmaxi[(size_t)dst[e] * NH + h];
  lm ^= (lm >> 31) & 0x7FFFFFFF;             // inverse transform (involution)
  float w = expf(logits[i] - __int_as_float(lm));
  wbuf[i] = w;
  unsafeAtomicAdd(&denom[(size_t)dst[e] * NH + h], w);
}

__global__ void att_alpha(const float* __restrict__ wbuf, const float* __restrict__ denom,
                          const int* __restrict__ dst, float* __restrict__ alpha) {
  int i = blockIdx.x * blockDim.x + threadIdx.x;
  if (i >= Ee * NH) return;
  int e = i >> 1, h = i & 1;
  alpha[i] = wbuf[i] / (denom[(size_t)dst[e] * NH + h] + 1e-16f);
}

// out[dst, h, c] += xl[src, h, c] * alpha[e, h]   (c over 2*128 = 256)
__global__ void att_accum(const float* __restrict__ xl, const float* __restrict__ alpha,
                          const int* __restrict__ src, const int* __restrict__ dst,
                          float* __restrict__ out) {
  int i = blockIdx.x * blockDim.x + threadIdx.x;
  if (i >= Ee * (NH * HID)) return;
  int e = i >> 8, c = i & 255;
  int h = c >> 7;
  unsafeAtomicAdd(&out[(size_t)dst[e] * (NH * HID) + c],
                  xl[(size_t)src[e] * (NH * HID) + c] * alpha[e * NH + h]);
}

// atten_out = relu(mean_heads + bias)
__global__ void head_mean_bias_relu(const float* __restrict__ acc,
                                    const float* __restrict__ bias,
                                    float* __restrict__ out) {
  int i = blockIdx.x * blockDim.x + threadIdx.x;
  if (i >= Nn * HID) return;
  int n = i >> 7, c = i & 127;
  float v = 0.5f * (acc[(size_t)n * 256 + c] + acc[(size_t)n * 256 + 128 + c]) + bias[c];
  out[i] = v > 0.f ? v : 0.f;
}

// ---------------- pooling + prediction head ----------------
__global__ void pool_add(const float* __restrict__ x, const int* __restrict__ bidx,
                         float* __restrict__ pool, int total, int C) {
  int i = blockIdx.x * blockDim.x + threadIdx.x;
  if (i >= total) return;
  int n = i / C, c = i - n * C;
  unsafeAtomicAdd(&pool[(size_t)bidx[n] * C + c], x[i]);
}

// 64 blocks (graph g) x 32 threads (mid m): out_h then 1-wide output
__global__ void pred_head(const float* __restrict__ pool0, const float* __restrict__ poolL,
                          const float* __restrict__ w0, const float* __restrict__ b0,
                          const float* __restrict__ wL, const float* __restrict__ bL,
                          const float* __restrict__ ow, const float* __restrict__ ob,
                          float* __restrict__ outp) {
  const int g = blockIdx.x, m = threadIdx.x;
  float s = b0[m] + bL[m];
  for (int k = 0; k < F_IN; ++k) s += pool0[g * F_IN + k] * w0[k * MIDd + m];
  for (int k = 0; k < HID;  ++k) s += poolL[g * HID  + k] * wL[k * MIDd + m];
  s = s > 0.f ? s : 0.f;
  __shared__ float sh[MIDd];
  sh[m] = s * ow[m];
  __syncthreads();
  if (m == 0) {
    float o = ob[0];
    for (int t = 0; t < MIDd; ++t) o += sh[t];
    outp[g] = o;
  }
}

// ---------------------------------------------------------------------------
extern "C" void kernel_launch(void* const* d_in, const int* in_sizes, int n_in,
                              void* d_out, int out_size, void* d_ws, size_t ws_size,
                              hipStream_t stream) {
  // ---- inputs (setup_inputs dict order) ----
  const float* x       = (const float*)d_in[0];
  const int*   eidx    = (const int*)d_in[1];    // [2,E] row-major
  const int*   aeidx   = (const int*)d_in[2];
  const int*   bidx    = (const int*)d_in[3];
  const float* g0_w1   = (const float*)d_in[4];
  const float* g0_b1   = (const float*)d_in[5];
  const float* g0_bng  = (const float*)d_in[6];
  const float* g0_bnb  = (const float*)d_in[7];
  const float* g0_w2   = (const float*)d_in[8];
  const float* g0_b2   = (const float*)d_in[9];
  const float* g1_w1   = (const float*)d_in[10];
  const float* g1_b1   = (const float*)d_in[11];
  const float* g1_bng  = (const float*)d_in[12];
  const float* g1_bnb  = (const float*)d_in[13];
  const float* g1_w2   = (const float*)d_in[14];
  const float* g1_b2   = (const float*)d_in[15];
  const float* bn0_g   = (const float*)d_in[16];
  const float* bn0_b   = (const float*)d_in[17];
  const float* bn1_g   = (const float*)d_in[18];
  const float* bn1_b   = (const float*)d_in[19];
  const float* a1_wl   = (const float*)d_in[20];
  const float* a1_bl   = (const float*)d_in[21];
  const float* a1_wr   = (const float*)d_in[22];
  const float* a1_br   = (const float*)d_in[23];
  const float* a1_att  = (const float*)d_in[24];
  const float* a1_bias = (const float*)d_in[25];
  const float* a2_wl   = (const float*)d_in[26];
  const float* a2_bl   = (const float*)d_in[27];
  const float* a2_wr   = (const float*)d_in[28];
  const float* a2_br   = (const float*)d_in[29];
  const float* a2_att  = (const float*)d_in[30];
  const float* a2_bias = (const float*)d_in[31];
  const float* p0_w    = (const float*)d_in[32];
  const float* p0_b    = (const float*)d_in[33];
  const float* pL_w    = (const float*)d_in[34];
  const float* pL_b    = (const float*)d_in[35];
  const float* out_w   = (const float*)d_in[36];
  const float* out_b   = (const float*)d_in[37];

  const int* src1 = eidx;          const int* dst1 = eidx + Ee;
  const int* src2 = aeidx;         const int* dst2 = aeidx + Ee;

  // ---- outputs: [outputs(64) | atten_out(N*128) | alpha(E*2)] ----
  float* out_pred  = (float*)d_out;
  float* atten_out = out_pred + Gg;
  float* alpha_out = atten_out + (size_t)Nn * HID;

  // ---- workspace layout (floats) ----
  float* f = (float*)d_ws;
  size_t o = 0;
  float* bufA   = f + o; o += (size_t)Nn * 256;  // xl1, later xr2
  float* bufB   = f + o; o += (size_t)Nn * 256;  // xr1, later xl2
  float* bufE   = f + o; o += (size_t)Nn * 256;  // emb acc, later out2 acc
  float* bt     = f + o; o += (size_t)Nn * HID;  // t / hidden
  float* by     = f + o; o += (size_t)Nn * HID;  // gemm out
  float* bh1    = f + o; o += (size_t)Nn * HID;
  float* bh2    = f + o; o += (size_t)Nn * HID;
  float* mean   = f + o; o += 256;
  float* inv    = f + o; o += 256;
  int*   lmaxi  = (int*)(f + o); o += (size_t)Nn * NH;
  float* denom  = f + o; o += (size_t)Nn * NH;
  float* logitb = f + o; o += (size_t)Ee * NH;
  float* wbuf   = f + o; o += (size_t)Ee * NH;
  float* alpha1 = f + o; o += (size_t)Ee * NH;
  float* pool0  = f + o; o += (size_t)Gg * F_IN;
  float* poolL  = f + o; o += (size_t)Gg * HID;

  const int TB = 256;
  auto nb = [](long long n, int tb) { return (unsigned)((n + tb - 1) / tb); };
  auto ggrid = [](int M) { return dim3((unsigned)(M / 64), Nn / 16); };  // 16x64 strips

  // ================= GANConv 0 =================
  hipMemcpyAsync(bt, x, (size_t)Nn * F_IN * sizeof(float), hipMemcpyDeviceToDevice, stream);
  edge_gather_add<<<nb((long long)Ee * F_IN, TB), TB, 0, stream>>>(x, src1, dst1, bt, Ee * F_IN, F_IN);
  gemm_bias_wmma_f32<<<ggrid(HID), 32, 0, stream>>>(bt, g0_w1, g0_b1, by, F_IN, HID);
  bn_stats<<<HID, 256, 0, stream>>>(by, HID, mean, inv);
  bn_relu<<<nb((long long)Nn * HID, TB), TB, 0, stream>>>(by, mean, inv, g0_bng, g0_bnb, bt, Nn * HID, HID);
  gemm_bias_wmma_f32<<<ggrid(HID), 32, 0, stream>>>(bt, g0_w2, g0_b2, by, HID, HID);
  bn_stats<<<HID, 256, 0, stream>>>(by, HID, mean, inv);
  bn_relu<<<nb((long long)Nn * HID, TB), TB, 0, stream>>>(by, mean, inv, bn0_g, bn0_b, bh1, Nn * HID, HID);

  // ================= GANConv 1 =================
  hipMemcpyAsync(bt, bh1, (size_t)Nn * HID * sizeof(float), hipMemcpyDeviceToDevice, stream);
  edge_gather_add<<<nb((long long)Ee * HID, TB), TB, 0, stream>>>(bh1, src1, dst1, bt, Ee * HID, HID);
  gemm_bias_wmma_f32<<<ggrid(HID), 32, 0, stream>>>(bt, g1_w1, g1_b1, by, HID, HID);
  bn_stats<<<HID, 256, 0, stream>>>(by, HID, mean, inv);
  bn_relu<<<nb((long long)Nn * HID, TB), TB, 0, stream>>>(by, mean, inv, g1_bng, g1_bnb, bt, Nn * HID, HID);
  gemm_bias_wmma_f32<<<ggrid(HID), 32, 0, stream>>>(bt, g1_w2, g1_b2, by, HID, HID);
  bn_stats<<<HID, 256, 0, stream>>>(by, HID, mean, inv);
  bn_relu<<<nb((long long)Nn * HID, TB), TB, 0, stream>>>(by, mean, inv, bn1_g, bn1_b, bh2, Nn * HID, HID);

  // ================= ATTConv 1 (concat) =================
  gemm_bias_wmma_f32<<<ggrid(256), 32, 0, stream>>>(bh2, a1_wl, a1_bl, bufA, HID, 256);
  gemm_bias_wmma_f32<<<ggrid(256), 32, 0, stream>>>(bh2, a1_wr, a1_br, bufB, HID, 256);
  fill_i32<<<nb(Nn * NH, TB), TB, 0, stream>>>(lmaxi, INT_MIN, Nn * NH);
  att_logits<<<nb((long long)Ee * NH, TB), TB, 0, stream>>>(bufA, bufB, src1, dst1, a1_att, logitb, lmaxi);
  hipMemsetAsync(denom, 0, (size_t)Nn * NH * sizeof(float), stream);
  att_expw<<<nb((long long)Ee * NH, TB), TB, 0, stream>>>(logitb, lmaxi, dst1, wbuf, denom);
  att_alpha<<<nb((long long)Ee * NH, TB), TB, 0, stream>>>(wbuf, denom, dst1, alpha1);
  hipMemsetAsync(bufE, 0, (size_t)Nn * 256 * sizeof(float), stream);
  att_accum<<<nb((long long)Ee * 256, TB), TB, 0, stream>>>(bufA, alpha1, src1, dst1, bufE);
  bias_relu<<<nb((long long)Nn * 256, TB), TB, 0, stream>>>(bufE, a1_bias, bufE, Nn * 256, 256);  // emb

  // ================= ATTConv 2 (mean over heads) =================
  gemm_bias_wmma_f32<<<ggrid(256), 32, 0, stream>>>(bufE, a2_wl, a2_bl, bufB, 256, 256); // xl2
  gemm_bias_wmma_f32<<<ggrid(256), 32, 0, stream>>>(bufE, a2_wr, a2_br, bufA, 256, 256); // xr2
  fill_i32<<<nb(Nn * NH, TB), TB, 0, stream>>>(lmaxi, INT_MIN, Nn * NH);
  att_logits<<<nb((long long)Ee * NH, TB), TB, 0, stream>>>(bufB, bufA, src2, dst2, a2_att, logitb, lmaxi);
  hipMemsetAsync(denom, 0, (size_t)Nn * NH * sizeof(float), stream);
  att_expw<<<nb((long long)Ee * NH, TB), TB, 0, stream>>>(logitb, lmaxi, dst2, wbuf, denom);
  att_alpha<<<nb((long long)Ee * NH, TB), TB, 0, stream>>>(wbuf, denom, dst2, alpha_out);      // output alpha
  hipMemsetAsync(bufE, 0, (size_t)Nn * 256 * sizeof(float), stream);
  att_accum<<<nb((long long)Ee * 256, TB), TB, 0, stream>>>(bufB, alpha_out, src2, dst2, bufE);
  head_mean_bias_relu<<<nb((long long)Nn * HID, TB), TB, 0, stream>>>(bufE, a2_bias, atten_out);

  // ================= pooling + prediction head =================
  hipMemsetAsync(pool0, 0, (size_t)Gg * F_IN * sizeof(float), stream);
  hipMemsetAsync(poolL, 0, (size_t)Gg * HID * sizeof(float), stream);
  pool_add<<<nb((long long)Nn * F_IN, TB), TB, 0, stream>>>(x, bidx, pool0, Nn * F_IN, F_IN);
  pool_add<<<nb((long long)Nn * HID, TB), TB, 0, stream>>>(atten_out, bidx, poolL, Nn * HID, HID);
  pred_head<<<Gg, MIDd, 0, stream>>>(pool0, poolL, p0_w, p0_b, pL_w, pL_b, out_w, out_b, out_pred);
}